// SGConv_54279796686854
// MI455X (gfx1250) — compile-verified
//
#include <hip/hip_runtime.h>

typedef __attribute__((ext_vector_type(2))) float v2f;
typedef __attribute__((ext_vector_type(4))) float v4f;
typedef __attribute__((ext_vector_type(8))) float v8f;

#define B_  4
#define C_  96
#define L_  32
#define H_  56
#define W_  56
#define HW_ (H_ * W_)
#define LHW_ ((size_t)L_ * H_ * W_)
#define SW_  68   // row stride for 64-wide LDS tiles: 68 = 4 mod 64 -> conflict-free b64 pairs
#define SL_  36   // row stride for 32-wide transposed tile in conv_l (also 4 mod 64... 36%64=36; 36*D==+-2 mod 64 unsolvable)

// ---------------------------------------------------------------------------
// Pass 0: build normalized length kernel klen[C][32] into workspace.
// ---------------------------------------------------------------------------
__global__ void build_klen(const float* __restrict__ k0,
                           const float* __restrict__ k1,
                           const float* __restrict__ decay,
                           float* __restrict__ klen) {
    int c = blockIdx.x * blockDim.x + threadIdx.x;
    if (c >= C_) return;
    float d = decay[c];
    float k[32];
#pragma unroll
    for (int t = 0; t < 16; ++t) k[t] = k0[c * 16 + t] * d;   // decay^(NUM_SCALES-1)
#pragma unroll
    for (int j = 0; j < 16; ++j) {                            // linear interp, scale=2
        float src = 0.5f * (float)j - 0.25f;
        src = src < 0.0f ? 0.0f : src;
        int lo = (int)src;
        float frac = src - (float)lo;
        int hi = lo + 1; if (hi > 15) hi = 15;
        k[16 + j] = k1[c * 16 + lo] * (1.0f - frac) + k1[c * 16 + hi] * frac;
    }
    float s = 0.0f;
#pragma unroll
    for (int t = 0; t < 32; ++t) s += k[t] * k[t];
    float inv = 1.0f / sqrtf(s);
#pragma unroll
    for (int t = 0; t < 32; ++t) klen[c * 32 + t] = k[t] * inv;
}

__device__ __forceinline__ float toep_tap(const float* __restrict__ g, int d, int len) {
    return (d >= 0 && d < len) ? g[d] : 0.0f;
}

// ---------------------------------------------------------------------------
// Pass 1: fused causal conv along W then H for one (b,c,l) slice.
//   Y1 = X * U_w ; Y2 = A_h * Y1.
//   Toeplitz taps pre-paired in LDS as float2 -> fragments are single b64 loads.
//   Y1 stored transposed (YsT[n][r]) so stage-2 B pairs are contiguous b64.
// ---------------------------------------------------------------------------
__global__ void __launch_bounds__(256)
conv_wh(const float* __restrict__ x, const float* __restrict__ wconv,
        const float* __restrict__ hconv, float* __restrict__ out) {
    __shared__ float Xs [64 * SW_];   // row-major input tile
    __shared__ float YsT[64 * SW_];   // transposed stage-1 output: YsT[n*SW_+r]
    __shared__ v2f   pw[128];         // pw[u] = {kw[63-u], kw[62-u]} (0 outside band)
    __shared__ v2f   ph[128];         // ph[u] = {kh[63-u], kh[62-u]}

    int slice = blockIdx.x;                 // ((b*C + c)*L + l)
    int c = (slice / L_) % C_;
    const float* xg = x + (size_t)slice * HW_;
    float*       og = out + (size_t)slice * HW_;

    int tid = threadIdx.x;
    if (tid < 128) {
        const float* wr = wconv + c * 56;
        const float* hr = hconv + c * 56;
        int d0 = 63 - tid, d1 = 62 - tid;
        v2f w; w.x = toep_tap(wr, d0, 56); w.y = toep_tap(wr, d1, 56);
        v2f h; h.x = toep_tap(hr, d0, 56); h.y = toep_tap(hr, d1, 56);
        pw[tid] = w;
        ph[tid] = h;
    }
    for (int i = tid; i < 64 * 64; i += 256) {
        int r = i >> 6, cc = i & 63;
        Xs[r * SW_ + cc] = (r < 56 && cc < 56) ? xg[r * W_ + cc] : 0.0f;
    }
    __syncthreads();

    int lane = tid & 31;
    int wave = __builtin_amdgcn_readfirstlane(tid >> 5);  // scalar wave id
    int m  = lane & 15;          // row-in-tile (A) / col-in-tile (B,C,D)
    int ko = (lane >> 4) * 2;    // K sub-offset {0,2}
    int rb = (lane >> 4) * 8;    // C/D row base offset {0,8}

    // ---- stage 1: Y1 = X * U_w ----
#pragma unroll
    for (int tt = 0; tt < 2; ++tt) {
        int t  = wave + tt * 8;             // scalar
        int ti = t >> 2, tj = t & 3;        // scalar
        int j  = tj * 16 + m;
        v8f acc = {};
        int nkt = tj + 1;                   // scalar triangular K-tile count
        for (int kt = 0; kt < nkt; ++kt) {
#pragma unroll
            for (int ks = 0; ks < 16; ks += 4) {
                int k0 = kt * 16 + ks + ko;
                v2f a = *(const v2f*)&Xs[(ti * 16 + m) * SW_ + k0];
                v2f b = pw[63 - j + k0];    // {kw[j-k0], kw[j-k0-1]}
                acc = __builtin_amdgcn_wmma_f32_16x16x4_f32(false, a, false, b,
                                                            (short)0, acc, false, false);
            }
        }
        // store transposed: YsT[j][row..row+7], contiguous per lane
        float* yp = &YsT[j * SW_ + ti * 16 + rb];
        v4f lo, hi;
        lo.x = acc[0]; lo.y = acc[1]; lo.z = acc[2]; lo.w = acc[3];
        hi.x = acc[4]; hi.y = acc[5]; hi.z = acc[6]; hi.w = acc[7];
        *(v4f*)(yp)     = lo;
        *(v4f*)(yp + 4) = hi;
    }
    __syncthreads();

    // ---- stage 2: Y2 = A_h * Y1 ----
#pragma unroll
    for (int tt = 0; tt < 2; ++tt) {
        int t  = wave + tt * 8;
        int ti = t >> 2, tj = t & 3;
        int i0 = ti * 16 + m;               // output row for A fragment
        int n  = tj * 16 + m;               // output col for B fragment
        v8f acc = {};
        int nkt = ti + 1;                   // scalar triangular K-tile count
        for (int kt = 0; kt < nkt; ++kt) {
#pragma unroll
            for (int ks = 0; ks < 16; ks += 4) {
                int k0 = kt * 16 + ks + ko;
                v2f a = ph[63 - i0 + k0];   // {kh[i0-k0], kh[i0-k0-1]}
                v2f b = *(const v2f*)&YsT[n * SW_ + k0];  // {Y1[k0][n], Y1[k0+1][n]}
                acc = __builtin_amdgcn_wmma_f32_16x16x4_f32(false, a, false, b,
                                                            (short)0, acc, false, false);
            }
        }
        int row = ti * 16 + rb;
#pragma unroll
        for (int i = 0; i < 8; ++i) {
            int r = row + i;
            if (r < 56 && n < 56) og[r * W_ + n] = acc[i];
        }
    }
}

// ---------------------------------------------------------------------------
// Pass 2: causal conv along L for one (b,c,h) slab, in place on d_out.
//   Y = A_l * X.  X staged transposed (XsT[n][l]) -> B pairs contiguous b64.
// ---------------------------------------------------------------------------
__global__ void __launch_bounds__(256)
conv_l(float* __restrict__ io, const float* __restrict__ klen) {
    __shared__ float XsT[64 * SL_];  // XsT[n*SL_+l]
    __shared__ v2f   pl[64];         // pl[u] = {kl[31-u], kl[30-u]}

    int slice = blockIdx.x;                 // (b*C + c)*H + h
    int h  = slice % H_;
    int bc = slice / H_;
    int c  = bc % C_;
    float* g = io + (size_t)bc * LHW_ + (size_t)h * W_;

    int tid = threadIdx.x;
    if (tid < 64) {
        const float* kr = klen + c * 32;
        v2f v; v.x = toep_tap(kr, 31 - tid, 32); v.y = toep_tap(kr, 30 - tid, 32);
        pl[tid] = v;
    }
    for (int i = tid; i < 32 * 64; i += 256) {
        int r = i >> 6, cc = i & 63;
        XsT[cc * SL_ + r] = (cc < 56) ? g[(size_t)r * HW_ + cc] : 0.0f;
    }
    __syncthreads();

    int lane = tid & 31;
    int wave = __builtin_amdgcn_readfirstlane(tid >> 5);
    int m  = lane & 15;
    int ko = (lane >> 4) * 2;
    int rb = (lane >> 4) * 8;
    int ti = wave >> 2, tj = wave & 3;      // scalar; 2x4 tile grid over 32x64

    int i0 = ti * 16 + m;
    int n  = tj * 16 + m;
    v8f acc = {};
    int nkt = ti + 1;                       // scalar triangular K-tile count
    for (int kt = 0; kt < nkt; ++kt) {
#pragma unroll
        for (int ks = 0; ks < 16; ks += 4) {
            int k0 = kt * 16 + ks + ko;
            v2f a = pl[31 - i0 + k0];       // {kl[i0-k0], kl[i0-k0-1]}
            v2f b = *(const v2f*)&XsT[n * SL_ + k0];  // {X[k0][n], X[k0+1][n]}
            acc = __builtin_amdgcn_wmma_f32_16x16x4_f32(false, a, false, b,
                                                        (short)0, acc, false, false);
        }
    }
    int row = ti * 16 + rb;
#pragma unroll
    for (int i = 0; i < 8; ++i) {
        if (n < 56) g[(size_t)(row + i) * HW_ + n] = acc[i];
    }
}

// ---------------------------------------------------------------------------
extern "C" void kernel_launch(void* const* d_in, const int* in_sizes, int n_in,
                              void* d_out, int out_size, void* d_ws, size_t ws_size,
                              hipStream_t stream) {
    (void)in_sizes; (void)n_in; (void)out_size; (void)ws_size;
    const float* x     = (const float*)d_in[0];   // (B,C,L,H,W)
    const float* wconv = (const float*)d_in[1];   // (C,W)
    const float* hconv = (const float*)d_in[2];   // (C,H)
    const float* k0    = (const float*)d_in[3];   // (1,C,16)
    const float* k1    = (const float*)d_in[4];   // (1,C,16)
    const float* decay = (const float*)d_in[5];   // (1,C,1)
    float* out  = (float*)d_out;
    float* klen = (float*)d_ws;                   // C*32 floats

    build_klen<<<1, 128, 0, stream>>>(k0, k1, decay, klen);
    conv_wh<<<B_ * C_ * L_, 256, 0, stream>>>(x, wconv, hconv, out);
    conv_l<<<B_ * C_ * H_, 256, 0, stream>>>(out, klen);
}